// TurbineGNN_44865228374583
// MI455X (gfx1250) — compile-verified
//
#include <hip/hip_runtime.h>
#include <hip/hip_bf16.h>

typedef __attribute__((ext_vector_type(2))) float v2f;
typedef __attribute__((ext_vector_type(8))) float v8f;

// Problem constants (shapes are fixed by the reference; N/E re-derived at launch).
#define IN_F 128
#define HID  64
#define OUT3 32

// ---------------------------------------------------------------------------
// Small elementwise helpers
// ---------------------------------------------------------------------------
__global__ void fill_kernel(float* __restrict__ p, float v, unsigned n) {
    unsigned i = blockIdx.x * blockDim.x + threadIdx.x;
    if (i < n) p[i] = v;
}

__global__ void deg_count_kernel(const int* __restrict__ dst, float* __restrict__ deg,
                                 unsigned E) {
    unsigned e = blockIdx.x * blockDim.x + threadIdx.x;
    if (e < E) atomicAdd(&deg[dst[e]], 1.0f);
}

__global__ void rsqrt_kernel(float* __restrict__ d, unsigned n) {
    unsigned i = blockIdx.x * blockDim.x + threadIdx.x;
    if (i < n) d[i] = rsqrtf(d[i]);
}

__global__ void norm_kernel(const int* __restrict__ src, const int* __restrict__ dst,
                            const float* __restrict__ dinv, float* __restrict__ nrm,
                            unsigned E) {
    unsigned e = blockIdx.x * blockDim.x + threadIdx.x;
    if (e < E) nrm[e] = dinv[src[e]] * dinv[dst[e]];
}

// ---------------------------------------------------------------------------
// Dense GEMM  H[M x FOUT] = A[M x K] @ W[K x FOUT]   (fp32 WMMA 16x16x4)
// One wave per 16x16 output tile; wave w of a block covers column tile w.
// Fragment layouts per CDNA5 ISA 7.12.2:
//   A 16x4 f32 : lane&15 = M ; VGPR v, lane-half h -> K = h*2 + v
//   B 4x16 f32 : lane&15 = N ; VGPR v, lane-half h -> K = h*2 + v
//   D 16x16 f32: VGPR r -> M = r (lanes 0-15) / r+8 (lanes 16-31), N = lane&15
// nrows is a multiple of 16 in this problem, so EXEC is all-ones through WMMA.
// ---------------------------------------------------------------------------
template <int K, int FOUT>
__global__ void gemm_wmma_kernel(const float* __restrict__ A,
                                 const float* __restrict__ W,
                                 float* __restrict__ H, int nrows) {
    const int lane    = threadIdx.x & 31;
    const int colTile = threadIdx.x >> 5;
    const int col     = colTile * 16 + (lane & 15);
    int row = blockIdx.x * 16 + (lane & 15);
    if (row >= nrows) row = nrows - 1;        // clamp loads; keeps EXEC full
    const int kk = (lane >> 4) << 1;          // 0 for lanes 0-15, 2 for 16-31

    v8f acc = {};
#pragma unroll
    for (int k0 = 0; k0 < K; k0 += 4) {
        v2f a, b;
        a.x = A[(size_t)row * K + (k0 + kk)];
        a.y = A[(size_t)row * K + (k0 + kk + 1)];
        b.x = W[(size_t)(k0 + kk) * FOUT + col];
        b.y = W[(size_t)(k0 + kk + 1) * FOUT + col];
        acc = __builtin_amdgcn_wmma_f32_16x16x4_f32(
            /*neg_a=*/false, a, /*neg_b=*/false, b,
            /*c_mod=*/(short)0, acc, /*reuse_a=*/false, /*reuse_b=*/false);
    }

    const int mbase = blockIdx.x * 16 + ((lane < 16) ? 0 : 8);
#pragma unroll
    for (int r = 0; r < 8; ++r) {
        int m = mbase + r;
        if (m < nrows) H[(size_t)m * FOUT + col] = acc[r];
    }
}

// ---------------------------------------------------------------------------
// Edge scatter: one thread per (edge, feature).
// agg[dst[e]][f] += h[src[e]][f] * norm[e]
// Feature tables (25.6 MB) + accumulator live in the 192 MB L2, so the random
// gather and the f32 atomics are L2 traffic; HBM only streams src/dst/norm.
// ---------------------------------------------------------------------------
__global__ void edge_agg_kernel(const int* __restrict__ src, const int* __restrict__ dst,
                                const float* __restrict__ nrm, const float* __restrict__ h,
                                float* __restrict__ agg, unsigned total, int logF) {
    unsigned gid = blockIdx.x * blockDim.x + threadIdx.x;
    if (gid >= total) return;
    unsigned e = gid >> logF;
    unsigned f = gid & ((1u << logF) - 1u);
    __builtin_prefetch(&nrm[e + 4096], 0, 1);     // global_prefetch_b8 on the stream
    float v = h[((size_t)src[e] << logF) + f] * nrm[e];
    atomicAdd(&agg[((size_t)dst[e] << logF) + f], v);
}

// out = relu(agg + h * dinv^2 + bias)   (self-loop folded in, no atomics needed)
__global__ void finish_kernel(const float* __restrict__ agg, const float* __restrict__ h,
                              const float* __restrict__ dinv, const float* __restrict__ bias,
                              float* __restrict__ out, unsigned total, int logF) {
    unsigned gid = blockIdx.x * blockDim.x + threadIdx.x;
    if (gid >= total) return;
    unsigned i = gid >> logF;
    unsigned f = gid & ((1u << logF) - 1u);
    float di = dinv[i];
    float v  = agg[gid] + h[gid] * di * di + bias[f];
    out[gid] = fmaxf(v, 0.0f);
}

// Final projection: wave per node, 32 lanes = 32 features, shuffle reduce.
__global__ void proj_kernel(const float* __restrict__ h, const float* __restrict__ Wp,
                            const float* __restrict__ bp, float* __restrict__ out,
                            unsigned n) {
    unsigned wid  = (blockIdx.x * blockDim.x + threadIdx.x) >> 5;
    unsigned lane = threadIdx.x & 31;
    if (wid >= n) return;
    float v = h[(size_t)wid * OUT3 + lane] * Wp[lane];
#pragma unroll
    for (int o = 16; o > 0; o >>= 1) v += __shfl_xor(v, o, 32);
    if (lane == 0) out[wid] = v + bp[0];
}

// ---------------------------------------------------------------------------
static inline size_t align256(size_t x) { return (x + 255u) & ~(size_t)255u; }

extern "C" void kernel_launch(void* const* d_in, const int* in_sizes, int n_in,
                              void* d_out, int out_size, void* d_ws, size_t ws_size,
                              hipStream_t stream) {
    const float* x  = (const float*)d_in[0];
    const int*   ei = (const int*)d_in[1];      // [2, E] flat: src row then dst row
    const float* W1 = (const float*)d_in[2];
    const float* b1 = (const float*)d_in[3];
    const float* W2 = (const float*)d_in[4];
    const float* b2 = (const float*)d_in[5];
    const float* W3 = (const float*)d_in[6];
    const float* b3 = (const float*)d_in[7];
    const float* Wp = (const float*)d_in[8];
    const float* bp = (const float*)d_in[9];
    float* out = (float*)d_out;

    const unsigned N = (unsigned)(in_sizes[0] / IN_F);
    const unsigned E = (unsigned)(in_sizes[1] / 2);
    const int* src = ei;
    const int* dst = ei + E;

    // Workspace carve-out
    char*  ws   = (char*)d_ws;
    size_t off  = 0;
    float* dinv = (float*)(ws + off); off += align256((size_t)N * 4);            // deg -> dinv
    float* nrm  = (float*)(ws + off); off += align256((size_t)E * 4);
    float* hA   = (float*)(ws + off); off += align256((size_t)N * HID * 4);
    float* hB   = (float*)(ws + off); off += align256((size_t)N * HID * 4);
    float* agg  = (float*)(ws + off); off += align256((size_t)N * HID * 4);
    (void)ws_size;

    const int T = 256;
    #define GRID(n) ((unsigned)(((n) + T - 1) / T))

    // Degree (self-loop contributes the initial 1.0), then dinv = rsqrt(deg)
    fill_kernel<<<GRID(N), T, 0, stream>>>(dinv, 1.0f, N);
    deg_count_kernel<<<GRID(E), T, 0, stream>>>(dst, dinv, E);
    rsqrt_kernel<<<GRID(N), T, 0, stream>>>(dinv, N);
    norm_kernel<<<GRID(E), T, 0, stream>>>(src, dst, dinv, nrm, E);

    const unsigned mtiles = (N + 15) / 16;

    // ---- Layer 1: 128 -> 64 ----
    gemm_wmma_kernel<IN_F, HID><<<mtiles, 32 * (HID / 16), 0, stream>>>(x, W1, hA, (int)N);
    fill_kernel<<<GRID(N * HID), T, 0, stream>>>(agg, 0.0f, N * HID);
    edge_agg_kernel<<<GRID(E * HID), T, 0, stream>>>(src, dst, nrm, hA, agg, E * HID, 6);
    finish_kernel<<<GRID(N * HID), T, 0, stream>>>(agg, hA, dinv, b1, hB, N * HID, 6);

    // ---- Layer 2: 64 -> 64 ----
    gemm_wmma_kernel<HID, HID><<<mtiles, 32 * (HID / 16), 0, stream>>>(hB, W2, hA, (int)N);
    fill_kernel<<<GRID(N * HID), T, 0, stream>>>(agg, 0.0f, N * HID);
    edge_agg_kernel<<<GRID(E * HID), T, 0, stream>>>(src, dst, nrm, hA, agg, E * HID, 6);
    finish_kernel<<<GRID(N * HID), T, 0, stream>>>(agg, hA, dinv, b2, hB, N * HID, 6);

    // ---- Layer 3: 64 -> 32 ----
    gemm_wmma_kernel<HID, OUT3><<<mtiles, 32 * (OUT3 / 16), 0, stream>>>(hB, W3, hA, (int)N);
    fill_kernel<<<GRID(N * OUT3), T, 0, stream>>>(agg, 0.0f, N * OUT3);
    edge_agg_kernel<<<GRID(E * OUT3), T, 0, stream>>>(src, dst, nrm, hA, agg, E * OUT3, 5);
    finish_kernel<<<GRID(N * OUT3), T, 0, stream>>>(agg, hA, dinv, b3, hB, N * OUT3, 5);

    // ---- Projection: [N,32] @ [32,1] + bp ----
    proj_kernel<<<GRID(N * 32), T, 0, stream>>>(hB, Wp, bp, out, N);
    #undef GRID
    (void)n_in; (void)out_size;
}